// Our_TS2Vec_loss_61710090109447
// MI455X (gfx1250) — compile-verified
//
#include <hip/hip_runtime.h>
#include <math.h>

// CDNA5 / gfx1250, wave32. fp32 WMMA 16x16x4 for all gram matrices.
// Cross-lane reductions via DPP (VALU, no LDS): quad_perm xor1/xor2,
// row_half_mirror, row_mirror complete a 16-lane butterfly; the final
// lane0/lane16 combine uses v_readlane.

typedef __attribute__((ext_vector_type(2))) float v2f;
typedef __attribute__((ext_vector_type(8))) float v8f;

#define BB 8           // batch
#define CC 64          // channels (K dim)
#define ALPHA_F 0.5f
#define INV_TAU 10.0f  // 1/0.1
#define EPS_F 1e-5f
#define NEG_BIG (-3.0e38f)

// DPP controls
#define DPP_XOR1 0xB1   // quad_perm [1,0,3,2]
#define DPP_XOR2 0x4E   // quad_perm [2,3,0,1]
#define DPP_HMIR 0x141  // row_half_mirror (combines quads within 8)
#define DPP_MIR  0x140  // row_mirror      (combines 8-groups within 16)

template <int CTRL>
__device__ __forceinline__ float dppmov(float v) {
  return __int_as_float(
      __builtin_amdgcn_update_dpp(0, __float_as_int(v), CTRL, 0xf, 0xf, true));
}

// butterfly reductions over each 16-lane half (result replicated in all 16 lanes)
__device__ __forceinline__ float redmax16(float v) {
  v = fmaxf(v, dppmov<DPP_XOR1>(v));
  v = fmaxf(v, dppmov<DPP_XOR2>(v));
  v = fmaxf(v, dppmov<DPP_HMIR>(v));
  v = fmaxf(v, dppmov<DPP_MIR>(v));
  return v;
}
__device__ __forceinline__ float redsum16(float v) {
  v += dppmov<DPP_XOR1>(v);
  v += dppmov<DPP_XOR2>(v);
  v += dppmov<DPP_HMIR>(v);
  v += dppmov<DPP_MIR>(v);
  return v;
}
// full-wave sum: 16-lane sums, then combine the two halves via readlane
__device__ __forceinline__ float redsum32_lane0(float v) {
  v = redsum16(v);
  float a = __int_as_float(__builtin_amdgcn_readlane(__float_as_int(v), 0));
  float b = __int_as_float(__builtin_amdgcn_readlane(__float_as_int(v), 16));
  return a + b;   // valid in every lane; use from lane 0
}

// A (and, by gram symmetry, B) fragment for V_WMMA_F32_16X16X4_F32, k-step s:
// lane l: row = row0 + (l&15), k = 4*s + 2*(l>>4) + {0,1}   (32-bit A 16x4 layout)
__device__ __forceinline__ v2f frag_rows(const float* __restrict__ Z, int row0, int s,
                                         int lane, int nrows) {
  int r = row0 + (lane & 15);
  int k = 4 * s + 2 * (lane >> 4);
  v2f f;
  if (r < nrows) { f.x = Z[(size_t)r * CC + k]; f.y = Z[(size_t)r * CC + k + 1]; }
  else           { f.x = 0.f; f.y = 0.f; }
  return f;
}

// ---------- init ----------
__global__ void k_init(float* out) {
  if (threadIdx.x == 0 && blockIdx.x == 0) out[0] = 0.f;
}

// ---------- per-series softmax constant: const_b = max_x ||z_bx||^2 / tau ----------
// (== whole-matrix max of scaled gram, exactly, by Cauchy-Schwarz)
__global__ void k_const(const float* __restrict__ z1, const float* __restrict__ z2,
                        int T, float* __restrict__ constB) {
  int b = blockIdx.x;
  int tid = threadIdx.x;
  const float* Z = (b < BB) ? (z1 + (size_t)b * T * CC)
                            : (z2 + (size_t)(b - BB) * T * CC);
  float mx = -1e30f;
  for (int x = tid; x < T; x += 256) {
    const float* row = Z + (size_t)x * CC;
    float s = 0.f;
#pragma unroll 16
    for (int c = 0; c < CC; ++c) s += row[c] * row[c];
    mx = fmaxf(mx, s * INV_TAU);
  }
  __shared__ float sh[256];
  sh[tid] = mx; __syncthreads();
  for (int s = 128; s > 0; s >>= 1) { if (tid < s) sh[tid] = fmaxf(sh[tid], sh[tid + s]); __syncthreads(); }
  if (tid == 0) constB[b] = sh[0];
}

// ---------- max-pool over time (kernel=2, stride=2) for both tensors ----------
__global__ void k_pool(const float* __restrict__ s1, const float* __restrict__ s2,
                       float* __restrict__ d1, float* __restrict__ d2, int Th, int Tin) {
  int i = blockIdx.x * blockDim.x + threadIdx.x;
  int total = BB * Th * CC;
  if (i >= total) return;
  int c = i & (CC - 1);
  int t = (i / CC) % Th;
  int b = i / (Th * CC);
  size_t j = ((size_t)b * Tin + 2 * t) * CC + c;
  d1[i] = fmaxf(s1[j], s1[j + CC]);
  d2[i] = fmaxf(s2[j], s2[j + CC]);
}

// ---------- instance loss: one wave per timestep, 16x16 gram via 16 WMMA f32 16x16x4 ----------
__global__ void k_instance(const float* __restrict__ z1, const float* __restrict__ z2,
                           int T, float coef, float* __restrict__ out) {
  int lane = threadIdx.x & 31;
  int wid = threadIdx.x >> 5;
  int t = blockIdx.x * 8 + wid;           // blockDim=256 -> 8 waves
  if (t >= T) return;                     // wave-uniform, EXEC stays all-ones inside

  int m = lane & 15;                      // fragment row = concat row
  const float* rowp = (m < 8) ? (z1 + ((size_t)m * T + t) * CC)
                              : (z2 + ((size_t)(m - 8) * T + t) * CC);
  v2f a[16];
  int kb = 2 * (lane >> 4);
#pragma unroll
  for (int s = 0; s < 16; ++s) { a[s].x = rowp[4 * s + kb]; a[s].y = rowp[4 * s + kb + 1]; }

  v8f cc = {};
#pragma unroll
  for (int s = 0; s < 16; ++s)
    cc = __builtin_amdgcn_wmma_f32_16x16x4_f32(false, a[s], false, a[s], (short)0, cc, false, false);

  // C/D layout: lane l, vgpr r -> sim[8*(l>>4)+r][l&15]
  int h = lane >> 4, n = lane & 15;
  float acc = 0.f;
#pragma unroll
  for (int r = 0; r < 8; ++r) {
    int mr = 8 * h + r;
    float v = cc[r];
    float rmax = redmax16((n == mr) ? NEG_BIG : v);          // exclude diagonal
    float se = redsum16((n == mr) ? 0.f : expf(v - rmax));
    float lse = rmax + logf(se);
    int pos = (mr < 8) ? mr + 8 : mr - 8;                    // positive pair column
    if (n == pos) acc += (lse - v);
  }
  float tot = redsum32_lane0(acc);
  if (lane == 0) atomicAdd(out, coef * tot);
}

// ---------- temporal pass A: one wave per (b, x-tile); stream y-tiles >= x-tile ----------
// Per-lane accumulation only in the hot loop; one 16-lane reduce per row at the end.
__global__ void k_tempA(const float* __restrict__ z1, const float* __restrict__ z2,
                        int T, int ntiles, const float* __restrict__ constB,
                        float* __restrict__ rowD, float* __restrict__ psum) {
  int b = blockIdx.y;                    // 0..15 (concat batch)
  int xt = blockIdx.x;
  int lane = threadIdx.x & 31;
  const float* Z = (b < BB) ? (z1 + (size_t)b * T * CC)
                            : (z2 + (size_t)(b - BB) * T * CC);
  float cb = constB[b];
  int x0 = xt * 16;
  v2f a[16];
#pragma unroll
  for (int s = 0; s < 16; ++s) a[s] = frag_rows(Z, x0, s, lane, T);

  float sexp[8];
#pragma unroll
  for (int r = 0; r < 8; ++r) sexp[r] = 0.f;
  float Sl = 0.f;                        // per-lane strict-upper sum of scaled
  int h = lane >> 4, n = lane & 15;

  for (int yt = xt; yt < ntiles; ++yt) {
    int y0 = yt * 16;
    v2f bf[16];
#pragma unroll
    for (int s = 0; s < 16; ++s) bf[s] = frag_rows(Z, y0, s, lane, T);
    v8f cc = {};
#pragma unroll
    for (int s = 0; s < 16; ++s)
      cc = __builtin_amdgcn_wmma_f32_16x16x4_f32(false, a[s], false, bf[s], (short)0, cc, false, false);

    int y = y0 + n;
#pragma unroll
    for (int r = 0; r < 8; ++r) {
      int x = x0 + 8 * h + r;
      float v = cc[r] * INV_TAU;                 // scaled = sim / tau
      bool up = (x < T) && (y < T) && (y > x);   // strict upper triangle
      Sl += up ? v : 0.f;
      sexp[r] += up ? __expf(v - cb) : 0.f;
    }
  }

#pragma unroll
  for (int r = 0; r < 8; ++r) {
    float se = redsum16(sexp[r]);                // denom[x] for row x = x0+8h+r
    int x = x0 + 8 * h + r;
    if (n == 0 && x < T) rowD[(size_t)b * T + x] = se;
  }
  float Sall = redsum32_lane0(Sl);
  if (lane == 0) psum[b * ntiles + xt] = Sall;
}

// ---------- temporal pass B: per-b reduce (weighted log-denoms, upper sum) ----------
__global__ void k_tempB(const float* __restrict__ rowD, const float* __restrict__ psum,
                        const float* __restrict__ constB,
                        int T, int ntiles, float coef, float* __restrict__ out) {
  int b = blockIdx.x;
  int tid = threadIdx.x;
  __shared__ float sh[256];
  float cb = constB[b];

  float wsum = 0.f;
  for (int x = tid; x < T; x += 256)
    wsum += (float)(T - 1 - x) * logf(rowD[(size_t)b * T + x] + EPS_F);
  float ssum = 0.f;
  for (int i = tid; i < ntiles; i += 256) ssum += psum[b * ntiles + i];

  sh[tid] = wsum; __syncthreads();
  for (int s = 128; s > 0; s >>= 1) { if (tid < s) sh[tid] += sh[tid + s]; __syncthreads(); }
  float W = sh[0]; __syncthreads();
  sh[tid] = ssum; __syncthreads();
  for (int s = 128; s > 0; s >>= 1) { if (tid < s) sh[tid] += sh[tid + s]; __syncthreads(); }
  float S = sh[0];

  if (tid == 0) {
    float NU = 0.5f * (float)T * (float)(T - 1);
    atomicAdd(out, coef * (NU * cb + W - S));
  }
}

extern "C" void kernel_launch(void* const* d_in, const int* in_sizes, int n_in,
                              void* d_out, int out_size, void* d_ws, size_t ws_size,
                              hipStream_t stream) {
  (void)n_in; (void)out_size; (void)ws_size;
  const float* z1 = (const float*)d_in[0];
  const float* z2 = (const float*)d_in[1];
  int T0 = in_sizes[0] / (BB * CC);

  // depth d (matches reference loop count)
  int dlev = 0; { int t = T0; while (t > 1) { dlev++; t >>= 1; } dlev++; }

  // workspace carve (floats): pool ping/pong + row denominators + partials + consts
  float* ws = (float*)d_ws;
  size_t T1 = (size_t)((T0 / 2 > 0) ? T0 / 2 : 1);
  size_t nA = 2ull * BB * T1 * CC;
  size_t T2 = (T1 / 2 > 0) ? T1 / 2 : 1;
  size_t nB = 2ull * BB * T2 * CC;
  float* bufA = ws;
  float* bufB = bufA + nA;
  float* rowD = bufB + nB;
  int nt0 = (T0 + 15) / 16;
  float* psum = rowD + (size_t)16 * T0;
  float* constB = psum + (size_t)16 * nt0;

  float* out = (float*)d_out;
  k_init<<<1, 32, 0, stream>>>(out);

  const float* c1 = z1;
  const float* c2 = z2;
  int T = T0, l = 0;
  while (true) {
    float coefI = ALPHA_F / (2.0f * BB * (float)T * (float)dlev);
    int instBlocks = (T + 7) / 8;
    k_instance<<<instBlocks, 256, 0, stream>>>(c1, c2, T, coefI, out);
    if (T == 1) break;

    int ntiles = (T + 15) / 16;
    k_const<<<16, 256, 0, stream>>>(c1, c2, T, constB);
    dim3 gA(ntiles, 16);
    k_tempA<<<gA, 32, 0, stream>>>(c1, c2, T, ntiles, constB, rowD, psum);
    float coefT = (1.0f - ALPHA_F) / (2.0f * BB * (float)dlev)
                  * 2.0f / ((float)T * (float)(T - 1));
    k_tempB<<<16, 256, 0, stream>>>(rowD, psum, constB, T, ntiles, coefT, out);

    int Th = T / 2;
    float* dst = ((l + 1) & 1) ? bufA : bufB;
    float* d1 = dst;
    float* d2 = dst + (size_t)BB * Th * CC;
    int total = BB * Th * CC;
    k_pool<<<(total + 255) / 256, 256, 0, stream>>>(c1, c2, d1, d2, Th, T);
    c1 = d1; c2 = d2; T = Th; ++l;
  }
}